// NMSPostProcess_48859547959903
// MI455X (gfx1250) — compile-verified
//
#include <hip/hip_runtime.h>
#include <hip/hip_bf16.h>
#include <math.h>

// ---------------- problem constants ----------------
#define BS      256
#define NQ      900
#define NC      91
#define NPB     (NQ*NC)          // 81900 logits per batch (divisible by 4)
#define K_TOP   100
#define IOU_THR 0.1f

// ---------------- kernel-1 tuning ----------------
#define T1      512              // 16 wave32s
#define CHUNK   4096             // floats per LDS staging buffer (16 KB)
#define NCH     ((NPB + CHUNK - 1)/CHUNK)   // 20 chunks
#define CAND    1024             // candidate cap for final sort

// output layout offsets (floats)
#define OFF_SCORES 0
#define OFF_BOXES  (BS*K_TOP)        // 25600
#define OFF_LABELS (BS*K_TOP*5)      // 128000
#define OFF_KEEP   (BS*K_TOP*6)      // 153600

// ---------------- CDNA5 async global->LDS helpers ----------------
#if __has_builtin(__builtin_amdgcn_s_wait_asynccnt)
#define WAIT_ASYNC(n) __builtin_amdgcn_s_wait_asynccnt(n)
#else
#define WAIT_ASYNC(n) asm volatile("s_wait_asynccnt %0" :: "i"(n) : "memory")
#endif

// One 16-byte async DMA: copies global memory at sbase + glob_off to the LDS byte address lds_byte, tracked by ASYNCcnt.
// VDST carries the (workgroup-relative) LDS byte address; generic shared pointers'
// low 32 bits are exactly that offset (aperture lives in the high 32 bits).
__device__ __forceinline__ void async_b128(unsigned lds_byte, unsigned glob_off,
                                           const float* sbase) {
  asm volatile("global_load_async_to_lds_b128 %0, %1, %2"
               :: "v"(lds_byte), "v"(glob_off), "s"(sbase) : "memory");
}

// Monotonic float -> uint key (ascending order preserved)
__device__ __forceinline__ unsigned fkey(float f) {
  unsigned u = __float_as_uint(f);
  return u ^ ((u & 0x80000000u) ? 0xFFFFFFFFu : 0x80000000u);
}

// Issue one chunk of async b128 copies. Every lane always issues (tail lanes
// clamp the *global* offset only), so each wave issues exactly 2 async
// instructions per chunk -> ASYNCcnt bookkeeping is exact.
__device__ __forceinline__ void issue_chunk(const float* lg, int c, float* dstbuf, int tid) {
  int nflt = NPB - c*CHUNK;
  int nvec = (nflt > CHUNK ? CHUNK : nflt) >> 2;     // float4 count (NPB%4==0)
  const unsigned gbase = (unsigned)c * (CHUNK*4u);
  const unsigned lbase = (unsigned)(unsigned long long)dstbuf;
#pragma unroll
  for (int r = 0; r < 2; ++r) {
    int v  = tid + r*T1;                 // 0..1023
    int gv = v < nvec ? v : nvec - 1;    // clamp (redundant copy, never read)
    async_b128(lbase + (unsigned)v*16u, gbase + (unsigned)gv*16u, lg);
  }
}

// =====================================================================
// Kernel 1: per-batch exact top-100 (radix-select + bitonic), box gather
// =====================================================================
__global__ __launch_bounds__(T1) void topk_kernel(
    const float* __restrict__ logits, const float* __restrict__ boxes,
    const int* __restrict__ tsizes, float* __restrict__ out)
{
  const int b   = blockIdx.x;
  const int tid = threadIdx.x;
  const float* lg = logits + (size_t)b * NPB;       // 16B aligned (81900*4*b % 16 == 0)

  __shared__ __align__(16) float stage[2][CHUNK];   // 32 KB double buffer
  __shared__ unsigned hist[2048];                   // 8 KB
  __shared__ unsigned long long cand[CAND];         // 8 KB
  __shared__ unsigned s_cnt, s_b1, s_above, s_thresh, s_skip;

  // ---- pass 1: HBM stream via async-to-LDS, histogram of top 11 key bits ----
  issue_chunk(lg, 0, stage[0], tid);                // prologue DMA overlaps zeroing
  for (int i = tid; i < 2048; i += T1) hist[i] = 0u;
  if (tid == 0) s_cnt = 0u;

  for (int c = 0; c < NCH; ++c) {
    const int buf = c & 1;
    if (c + 1 < NCH) {
      issue_chunk(lg, c + 1, stage[buf ^ 1], tid);  // outstanding: 4 per wave
      WAIT_ASYNC(2);                                // chunk c landed (in-order)
    } else {
      WAIT_ASYNC(0);
    }
    __syncthreads();                                // all waves' DMA visible
    int n_c = NPB - c*CHUNK; if (n_c > CHUNK) n_c = CHUNK;
    const float4* st4 = (const float4*)stage[buf];  // n_c % 4 == 0 always
    for (int i = tid; i < (n_c >> 2); i += T1) {    // ds_load_b128 consumption
      float4 v = st4[i];
      atomicAdd(&hist[fkey(v.x) >> 21], 1u);
      atomicAdd(&hist[fkey(v.y) >> 21], 1u);
      atomicAdd(&hist[fkey(v.z) >> 21], 1u);
      atomicAdd(&hist[fkey(v.w) >> 21], 1u);
    }
    __syncthreads();                                // before buffer reuse
  }

  if (tid == 0) {                                   // coarse bin from the top
    unsigned acc = 0u, b1 = 0u, inbin = 0u;
    for (int i = 2047; i >= 0; --i) {
      unsigned cnt = hist[i];
      if (acc + cnt >= K_TOP) { b1 = (unsigned)i; inbin = cnt; break; }
      acc += cnt;
    }
    // Common case: everything at/above the coarse bin fits the candidate
    // buffer -> skip the 11-bit refinement pass entirely (saves one full
    // L2 sweep of the logits).
    if (acc + inbin <= CAND) {
      s_thresh = b1 << 21;
      s_skip   = 1u;
    } else {
      s_b1 = b1; s_above = acc; s_skip = 0u;
    }
  }
  __syncthreads();

  if (s_skip == 0u) {                               // uniform branch per block
    // ---- pass 2 (L2-resident): refine bin B1 on next 11 key bits ----
    const unsigned B1 = s_b1, above = s_above;
    for (int i = tid; i < 2048; i += T1) hist[i] = 0u;
    __syncthreads();
    const float4* lg4r = (const float4*)lg;
    for (int i = tid; i < NPB/4; i += T1) {
      float4 v = lg4r[i];
      unsigned k0 = fkey(v.x), k1 = fkey(v.y), k2 = fkey(v.z), k3 = fkey(v.w);
      if ((k0 >> 21) == B1) atomicAdd(&hist[(k0 >> 10) & 2047u], 1u);
      if ((k1 >> 21) == B1) atomicAdd(&hist[(k1 >> 10) & 2047u], 1u);
      if ((k2 >> 21) == B1) atomicAdd(&hist[(k2 >> 10) & 2047u], 1u);
      if ((k3 >> 21) == B1) atomicAdd(&hist[(k3 >> 10) & 2047u], 1u);
    }
    __syncthreads();
    if (tid == 0) {
      unsigned need = K_TOP - above;
      unsigned acc = 0u, b2 = 0u;
      for (int i = 2047; i >= 0; --i) {
        unsigned cnt = hist[i];
        if (acc + cnt >= need) { b2 = (unsigned)i; break; }
        acc += cnt;
      }
      s_thresh = (B1 << 21) | (b2 << 10);           // key >= thresh -> candidate
    }
    __syncthreads();
  }
  const unsigned TH = s_thresh;

  // ---- collection pass (L2-resident) ----
  const float4* lg4 = (const float4*)lg;
  for (int i = tid; i < NPB/4; i += T1) {
    float4 v = lg4[i];
    float e[4] = {v.x, v.y, v.z, v.w};
#pragma unroll
    for (int c4 = 0; c4 < 4; ++c4) {
      unsigned k = fkey(e[c4]);
      if (k >= TH) {
        unsigned p = atomicAdd(&s_cnt, 1u);
        unsigned idx = (unsigned)(i*4 + c4);
        if (p < CAND)
          cand[p] = ((unsigned long long)k << 32) | (0xFFFFFFFFu - idx);
      }
    }
  }
  __syncthreads();
  unsigned cnt = s_cnt; if (cnt > CAND) cnt = CAND;
  for (int i = tid; i < CAND; i += T1) if (i >= (int)cnt) cand[i] = 0ull;
  __syncthreads();

  // ---- bitonic sort, descending by (key, ~index): deterministic top-100 ----
  for (unsigned kk = 2; kk <= CAND; kk <<= 1) {
    for (unsigned j = kk >> 1; j > 0; j >>= 1) {
      for (unsigned i = tid; i < CAND; i += T1) {
        unsigned ixj = i ^ j;
        if (ixj > i) {
          unsigned long long a = cand[i], c2 = cand[ixj];
          bool up = ((i & kk) == 0);                 // inverted -> descending sort
          if (up ? (a < c2) : (a > c2)) { cand[i] = c2; cand[ixj] = a; }
        }
      }
      __syncthreads();
    }
  }

  // ---- emit top-100: sigmoid on winners only, gather + scale boxes ----
  if (tid < K_TOP) {
    unsigned long long e = cand[tid];
    unsigned idx = (e == 0ull) ? 0u : (0xFFFFFFFFu - (unsigned)(e & 0xFFFFFFFFull));
    float logit = lg[idx];
    float score = 1.0f / (1.0f + expf(-logit));
    int q   = (int)(idx / NC);
    int lab = (int)(idx % NC);
    const float* bp = boxes + ((size_t)b*NQ + q)*4;
    float cx = bp[0], cy = bp[1], w = bp[2], h = bp[3];
    float img_h = (float)tsizes[b*2 + 0];
    float img_w = (float)tsizes[b*2 + 1];
    float* ob = out + OFF_BOXES + ((size_t)b*K_TOP + tid)*4;
    ob[0] = (cx - 0.5f*w)*img_w;
    ob[1] = (cy - 0.5f*h)*img_h;
    ob[2] = (cx + 0.5f*w)*img_w;
    ob[3] = (cy + 0.5f*h)*img_h;
    out[OFF_SCORES + b*K_TOP + tid] = score;
    out[OFF_LABELS + b*K_TOP + tid] = (float)lab;
  }
}

// =====================================================================
// Kernel 2: per-batch class-aware NMS over 100 boxes (tiny, LDS-resident)
// =====================================================================
#define T2 128
__global__ __launch_bounds__(T2) void nms_kernel(float* __restrict__ out)
{
  const int b = blockIdx.x, tid = threadIdx.x;
  float*       sc = out + OFF_SCORES + b*K_TOP;
  float*       bx = out + OFF_BOXES  + (size_t)b*K_TOP*4;
  const float* lb = out + OFF_LABELS + b*K_TOP;
  float*       kp = out + OFF_KEEP   + b*K_TOP;

  __shared__ float s_box[K_TOP*4];
  __shared__ float s_off[K_TOP*4];
  __shared__ float s_area[K_TOP];
  __shared__ float s_iou[K_TOP*K_TOP];   // 40 KB
  __shared__ float s_red[T2];
  __shared__ float s_maxc;
  __shared__ unsigned char s_sup[K_TOP];

  for (int i = tid; i < K_TOP*4; i += T2) s_box[i] = bx[i];
  if (tid < K_TOP) s_sup[tid] = 0;
  __syncthreads();

  float m = -3.4e38f;
  for (int i = tid; i < K_TOP*4; i += T2) m = fmaxf(m, s_box[i]);
  s_red[tid] = m;
  __syncthreads();
  if (tid == 0) {
    float mm = s_red[0];
    for (int i = 1; i < T2; ++i) mm = fmaxf(mm, s_red[i]);
    s_maxc = mm;
  }
  __syncthreads();

  const float offscale = s_maxc + 1.0f;
  if (tid < K_TOP) {
    float off = lb[tid] * offscale;     // class offset trick (reference)
    float x1 = s_box[tid*4+0] + off, y1 = s_box[tid*4+1] + off;
    float x2 = s_box[tid*4+2] + off, y2 = s_box[tid*4+3] + off;
    s_off[tid*4+0] = x1; s_off[tid*4+1] = y1;
    s_off[tid*4+2] = x2; s_off[tid*4+3] = y2;
    s_area[tid] = fmaxf(x2 - x1, 0.0f) * fmaxf(y2 - y1, 0.0f);
  }
  __syncthreads();

  for (int p = tid; p < K_TOP*K_TOP; p += T2) {
    int i = p / K_TOP, j = p - i*K_TOP;
    float ix1 = fmaxf(s_off[i*4+0], s_off[j*4+0]);
    float iy1 = fmaxf(s_off[i*4+1], s_off[j*4+1]);
    float ix2 = fminf(s_off[i*4+2], s_off[j*4+2]);
    float iy2 = fminf(s_off[i*4+3], s_off[j*4+3]);
    float inter = fmaxf(ix2 - ix1, 0.0f) * fmaxf(iy2 - iy1, 0.0f);
    float uni   = s_area[i] + s_area[j] - inter;
    s_iou[p] = inter / fmaxf(uni, 1e-9f);
  }
  __syncthreads();

  for (int i = 0; i < K_TOP; ++i) {
    bool alive = (s_sup[i] == 0);
    if (alive && tid < K_TOP && tid > i && s_iou[i*K_TOP + tid] > IOU_THR)
      s_sup[tid] = 1;
    __syncthreads();
  }

  if (tid < K_TOP) {
    bool keep = (s_sup[tid] == 0);
    kp[tid] = keep ? 1.0f : 0.0f;
    if (!keep) {
      sc[tid] = 0.0f;
      float* ob = bx + tid*4;
      ob[0] = 0.0f; ob[1] = 0.0f; ob[2] = 0.0f; ob[3] = 0.0f;
    }
  }
}

// =====================================================================
extern "C" void kernel_launch(void* const* d_in, const int* in_sizes, int n_in,
                              void* d_out, int out_size, void* d_ws, size_t ws_size,
                              hipStream_t stream) {
  (void)in_sizes; (void)n_in; (void)out_size; (void)d_ws; (void)ws_size;
  const float* logits = (const float*)d_in[0];   // (256,900,91) f32
  const float* boxes  = (const float*)d_in[1];   // (256,900,4)  f32
  const int*   tsizes = (const int*)d_in[2];     // (256,2)      i32
  float* out = (float*)d_out;                    // scores|boxes|labels|keep

  topk_kernel<<<BS, T1, 0, stream>>>(logits, boxes, tsizes, out);
  nms_kernel <<<BS, T2, 0, stream>>>(out);
}